// AdditiveAttention_48180943127038
// MI455X (gfx1250) — compile-verified
//
#include <hip/hip_runtime.h>
#include <hip/hip_bf16.h>

typedef float v2f __attribute__((ext_vector_type(2)));
typedef float v8f __attribute__((ext_vector_type(8)));

#if __has_builtin(__builtin_amdgcn_tanhf)
#define TANH(x) __builtin_amdgcn_tanhf(x)
#else
#define TANH(x) tanhf(x)
#endif

#define B_   8
#define QL   512
#define KL   512
#define H_   64
#define DV   64
#define LDSQ 68   // row stride (floats) for 16x64 tiles: 16B-aligned, bank-rotating
#define PSTR 18   // row stride for the 16x16 probability tile

// ---------------------------------------------------------------------------
// Kernel 1: projections  qp = queries @ Wq,  kp = keys @ Wk
// One wave32 computes one 16x16 f32 output tile via 16 chained
// V_WMMA_F32_16X16X4_F32 (K = 64 in steps of 4).
// grid = (512 row-tiles [first 256 = queries, rest = keys], 4 col-tiles)
// ---------------------------------------------------------------------------
__global__ __launch_bounds__(32)
void aa_proj_kernel(const float* __restrict__ queries,
                    const float* __restrict__ keys,
                    const float* __restrict__ Wq,
                    const float* __restrict__ Wk,
                    float* __restrict__ qp,
                    float* __restrict__ kp)
{
    int rt = blockIdx.x;          // 0..511
    int ct = blockIdx.y;          // 0..3
    const float* X; const float* W; float* O;
    if (rt < (B_*QL)/16) { X = queries; W = Wq; O = qp; }
    else                 { X = keys;    W = Wk; O = kp; rt -= (B_*QL)/16; }

    const int lane = threadIdx.x;
    const int hi   = lane >> 4;   // 0 or 1
    const int lo   = lane & 15;
    const int rowBase = rt * 16;
    const int colBase = ct * 16;

    v8f c = {};
#pragma unroll
    for (int k0 = 0; k0 < H_; k0 += 4) {
        v2f a, b;
        // A 16x4: lanes 0-15 -> K {k0,k0+1}, lanes 16-31 -> K {k0+2,k0+3}
        a.x = X[(rowBase + lo) * H_ + k0 + hi * 2 + 0];
        a.y = X[(rowBase + lo) * H_ + k0 + hi * 2 + 1];
        // B 4x16: mirrored K striping, N = lo
        b.x = W[(k0 + hi * 2 + 0) * H_ + colBase + lo];
        b.y = W[(k0 + hi * 2 + 1) * H_ + colBase + lo];
        c = __builtin_amdgcn_wmma_f32_16x16x4_f32(false, a, false, b,
                                                  (short)0, c, false, false);
    }
#pragma unroll
    for (int r = 0; r < 8; ++r) {
        int row = rowBase + r + hi * 8;      // D: VGPR r -> M=r (lo half), r+8 (hi half)
        O[row * H_ + colBase + lo] = c[r];
    }
}

// ---------------------------------------------------------------------------
// Kernel 2: fused additive-attention with online softmax.
// One wave32 owns one (batch, 16-query-row) tile; loops over 16-key tiles up
// to ceil(valid_len/16) (masked tail contributes exactly 0 after softmax).
// Scores via v_tanh_f32 VALU; P@V accumulated with WMMA f32 16x16x4.
// grid = (32 q-tiles, 8 batches)
// ---------------------------------------------------------------------------
__global__ __launch_bounds__(32)
void aa_attn_kernel(const float* __restrict__ qp,
                    const float* __restrict__ kp,
                    const float* __restrict__ values,
                    const int*   __restrict__ valid_lens,
                    const float* __restrict__ wv,
                    float* __restrict__ out)
{
    const int qt   = blockIdx.x;     // 0..31
    const int b    = blockIdx.y;     // 0..7
    const int lane = threadIdx.x;
    const int hi   = lane >> 4;
    const int lo   = lane & 15;

    __shared__ float qs[16 * LDSQ];
    __shared__ float ks[16 * LDSQ];
    __shared__ float vs[16 * LDSQ];
    __shared__ float ps[16 * PSTR];
    __shared__ float wvs[H_];
    __shared__ float m_s[16], l_s[16], rs_s[16];

    // stage the 16x64 query tile (float4, bank-rotated rows)
    const float* qbase = qp + (b * QL + qt * 16) * H_;
    for (int t = lane; t < 256; t += 32) {          // 256 float4s
        int row = t >> 4, c4 = t & 15;
        *(float4*)&qs[row * LDSQ + c4 * 4] =
            *(const float4*)&qbase[row * H_ + c4 * 4];
    }
    if (lane < 16) {
        *(float4*)&wvs[lane * 4] = *(const float4*)&wv[lane * 4];
        m_s[lane] = -1e30f;
        l_s[lane] = 0.0f;
    }
    const int vlen = valid_lens[b];
    const int nkt  = (vlen + 15) >> 4;              // early exit: masked tiles are 0
    v8f acc0 = {}, acc1 = {}, acc2 = {}, acc3 = {};
    __syncthreads();

    for (int kt = 0; kt < nkt; ++kt) {
        const float* kbase = kp     + (b * KL + kt * 16) * H_;
        const float* vbase = values + (b * KL + kt * 16) * DV;
        for (int t = lane; t < 256; t += 32) {
            int row = t >> 4, c4 = t & 15;
            *(float4*)&ks[row * LDSQ + c4 * 4] =
                *(const float4*)&kbase[row * H_ + c4 * 4];
            *(float4*)&vs[row * LDSQ + c4 * 4] =
                *(const float4*)&vbase[row * DV + c4 * 4];
        }
        __syncthreads();

        // 16x16 score tile: lane handles column kj = lo, rows qi = hi*8 .. hi*8+7
        const int kj = lo;
        const int kg = kt * 16 + kj;
#pragma unroll
        for (int i = 0; i < 8; ++i) {
            int qi = i + hi * 8;
            float s = 0.0f;
#pragma unroll 8
            for (int h = 0; h < H_; ++h)
                s = fmaf(wvs[h], TANH(qs[qi * LDSQ + h] + ks[kj * LDSQ + h]), s);
            ps[qi * PSTR + kj] = (kg < vlen) ? s : -1000000.0f;
        }
        __syncthreads();

        // online softmax row update (one row per lane 0-15)
        if (lane < 16) {
            int r = lane;
            float mo = m_s[r];
            float tmax = -1e30f;
#pragma unroll
            for (int c = 0; c < 16; ++c) tmax = fmaxf(tmax, ps[r * PSTR + c]);
            float mn = fmaxf(mo, tmax);
            float sc = __expf(mo - mn);
            float tsum = 0.0f;
#pragma unroll
            for (int c = 0; c < 16; ++c) {
                float p = __expf(ps[r * PSTR + c] - mn);   // masked -> exp(-1e6-mn)=0
                ps[r * PSTR + c] = p;
                tsum += p;
            }
            l_s[r]  = l_s[r] * sc + tsum;
            m_s[r]  = mn;
            rs_s[r] = sc;
        }
        __syncthreads();

        // rescale accumulators by per-row exp(m_old - m_new)
        float scv[8];
#pragma unroll
        for (int r = 0; r < 8; ++r) scv[r] = rs_s[r + hi * 8];
#pragma unroll
        for (int r = 0; r < 8; ++r) {
            acc0[r] *= scv[r]; acc1[r] *= scv[r];
            acc2[r] *= scv[r]; acc3[r] *= scv[r];
        }

        // acc += P(16x16) @ V(16x64): 4 K-steps x 4 N-tiles of WMMA f32 16x16x4
#pragma unroll
        for (int kk = 0; kk < 16; kk += 4) {
            v2f a, b0, b1, b2, b3;
            a.x = ps[lo * PSTR + kk + hi * 2 + 0];
            a.y = ps[lo * PSTR + kk + hi * 2 + 1];
            int kr0 = (kk + hi * 2 + 0) * LDSQ;
            int kr1 = (kk + hi * 2 + 1) * LDSQ;
            b0.x = vs[kr0 +  0 + lo]; b0.y = vs[kr1 +  0 + lo];
            b1.x = vs[kr0 + 16 + lo]; b1.y = vs[kr1 + 16 + lo];
            b2.x = vs[kr0 + 32 + lo]; b2.y = vs[kr1 + 32 + lo];
            b3.x = vs[kr0 + 48 + lo]; b3.y = vs[kr1 + 48 + lo];
            acc0 = __builtin_amdgcn_wmma_f32_16x16x4_f32(false, a, false, b0, (short)0, acc0, false, false);
            acc1 = __builtin_amdgcn_wmma_f32_16x16x4_f32(false, a, false, b1, (short)0, acc1, false, false);
            acc2 = __builtin_amdgcn_wmma_f32_16x16x4_f32(false, a, false, b2, (short)0, acc2, false, false);
            acc3 = __builtin_amdgcn_wmma_f32_16x16x4_f32(false, a, false, b3, (short)0, acc3, false, false);
        }
        __syncthreads();
    }

    // epilogue: divide by softmax denominator and store
    float linv[8];
#pragma unroll
    for (int r = 0; r < 8; ++r) linv[r] = 1.0f / l_s[r + hi * 8];
    float* obase = out + (b * QL + qt * 16) * DV;
#pragma unroll
    for (int r = 0; r < 8; ++r) {
        int row = r + hi * 8;
        obase[row * DV +  0 + lo] = acc0[r] * linv[r];
        obase[row * DV + 16 + lo] = acc1[r] * linv[r];
        obase[row * DV + 32 + lo] = acc2[r] * linv[r];
        obase[row * DV + 48 + lo] = acc3[r] * linv[r];
    }
}

extern "C" void kernel_launch(void* const* d_in, const int* in_sizes, int n_in,
                              void* d_out, int out_size, void* d_ws, size_t ws_size,
                              hipStream_t stream) {
    const float* queries    = (const float*)d_in[0];
    const float* keys       = (const float*)d_in[1];
    const float* values     = (const float*)d_in[2];
    const int*   valid_lens = (const int*)  d_in[3];
    const float* Wq         = (const float*)d_in[4];
    const float* Wk         = (const float*)d_in[5];
    const float* wv         = (const float*)d_in[6];
    float* out = (float*)d_out;

    float* qp = (float*)d_ws;                 // [B*Q, H]  = 1 MB
    float* kp = qp + B_ * QL * H_;            // [B*K, H]  = 1 MB

    aa_proj_kernel<<<dim3((2 * B_ * QL) / 16, H_ / 16), 32, 0, stream>>>(
        queries, keys, Wq, Wk, qp, kp);
    aa_attn_kernel<<<dim3(QL / 16, B_), 32, 0, stream>>>(
        qp, kp, values, valid_lens, wv, out);
}